// ModuleOptim_86861418594992
// MI455X (gfx1250) — compile-verified
//
#include <hip/hip_runtime.h>
#include <math.h>

// Problem constants (from reference): M=16, K=512, P=2048, L=1024,
// fs=16000, so=40, delay_max=0.02 -> dmax=320, D=400, taps=81, Q=P+L-1=3071
#define MM   16
#define KK   512
#define PP   2048
#define LL   1024
#define NROWS (MM * KK)        // 8192
#define SO   40
#define DD   400               // dmax + 2*so
#define NTAP 81                // 2*so + 1
#define QQ   (PP + LL - 1)     // 3071
#define PI_F 3.14159265358979323846f

// stage-3 tiling: each wave owns 16 rows x 128 cols (8 WMMA t-tiles)
#define NTILES  8
#define TG_COLS (16 * NTILES)  // 128
#define VPAD_LO 148            // covers min band index -145 (b.y at pa_max)
#define VPAD_SZ 1304           // covers max band index 1153 (+VPAD_LO)

typedef __attribute__((ext_vector_type(2))) float v2f;
typedef __attribute__((ext_vector_type(8))) float v8f;

// ---------------------------------------------------------------------------
// Stage 1: per-row window -> compact 81 taps + (d0, sign) meta.
//   H_est[p] = sum_{j=0..80} taps[j] * H[p - s*(d0+j)],  d0 = am-2so
// ---------------------------------------------------------------------------
__device__ __forceinline__ float win_val(int j, float fa, float ratio) {
    float T  = (float)(j - SO) - fa;
    float cw = 0.5f * (1.0f + cosf(PI_F * T / (float)SO));
    float pt = PI_F * T;
    float sc = (T == 0.0f) ? 1.0f : (sinf(pt) / pt);
    return ratio * cw * sc;
}

__global__ void stage1_taps(const float* __restrict__ U,
                            const float* __restrict__ toa,
                            float* __restrict__ taps,
                            int*   __restrict__ meta) {
    int n = blockIdx.x * blockDim.x + threadIdx.x;
    if (n >= NROWS) return;

    float th = tanhf(U[n]);
    if (isnan(th)) th = 0.0f;                  // nan_to_num
    float delta = 0.02f * th;
    float frac  = delta * 16000.0f;
    float fa    = fabsf(frac);
    float mt    = toa[n];
    float nt    = delta + mt;
    if (nt == 0.0f) nt = 1.0f;
    float ratio = mt / nt;

    float best = -INFINITY; int am = 0;
    for (int j = 0; j < DD; ++j) {
        float w = win_val(j, fa, ratio);
        if (w > best) { best = w; am = j; }
    }

    for (int j = 0; j < NTAP; ++j) {
        int wi = am - SO + j;
        float w = (wi >= 0 && wi < DD) ? win_val(wi, fa, ratio) : 0.0f;
        taps[n * NTAP + j] = w;
    }
    meta[2 * n]     = am - 2 * SO;             // d0
    meta[2 * n + 1] = (frac >= 0.0f) ? 1 : -1; // s
}

// ---------------------------------------------------------------------------
// Stage 2: 81-tap FIR per row -> H_est (2nd output slot of d_out)
// ---------------------------------------------------------------------------
__global__ void stage2_hest(const float* __restrict__ H,
                            const float* __restrict__ taps,
                            const int*   __restrict__ meta,
                            float* __restrict__ Hest) {
    __shared__ float st[NTAP];
    __shared__ int   sd0, ssg;
    int n = blockIdx.y;
    if (threadIdx.x < NTAP) st[threadIdx.x] = taps[n * NTAP + threadIdx.x];
    if (threadIdx.x == 0) { sd0 = meta[2 * n]; ssg = meta[2 * n + 1]; }
    __syncthreads();

    int p = blockIdx.x * blockDim.x + threadIdx.x;
    if (p >= PP) return;
    const float* h = H + (size_t)n * PP;
    int d0 = sd0, s = ssg;
    float acc = 0.0f;
    #pragma unroll 3
    for (int j = 0; j < NTAP; ++j) {
        int q = p - s * (d0 + j);
        float hv = (q >= 0 && q < PP) ? h[q] : 0.0f;
        acc = fmaf(st[j], hv, acc);
    }
    Hest[(size_t)n * PP + p] = acc;
}

// ---------------------------------------------------------------------------
// Stage 3: x(N x Q) = H_est(N x P) * Toeplitz(v)(P x Q), banded, via
// V_WMMA_F32_16X16X4_F32. One wave per 16-row x 128-col slab (8 WMMA tiles):
// the A fragment (depends only on p) feeds all 8 tiles; v is zero-padded in
// LDS so B loads need no guards; the p-loop main body is guard-free with an
// aligned float2 A load (tail is kept for safety but is unreachable since
// P % 4 == 0 and chunks are 4-aligned).
// ---------------------------------------------------------------------------
__global__ void __launch_bounds__(128)
stage3_conv_wmma(const float* __restrict__ Hest,
                 const float* __restrict__ v,
                 float* __restrict__ x) {
    __shared__ float vs[VPAD_SZ];
    for (int i = threadIdx.x; i < VPAD_SZ; i += blockDim.x) {
        int k = i - VPAD_LO;
        vs[i] = (k >= 0 && k < LL) ? v[k] : 0.0f;
    }
    __syncthreads();

    int wave = threadIdx.x >> 5;
    int lane = threadIdx.x & 31;
    int g    = blockIdx.x * 4 + wave;      // t-group 0..23 (128 cols each)
    int n0   = blockIdx.y * 16;
    int t0   = g * TG_COLS;

    int half = lane >> 4;                  // 0 or 1
    int lidx = lane & 15;                  // A row / B,D col
    int kb   = half * 2;                   // k offset within chunk of 4

    int pLo = t0 - (LL - 1); if (pLo < 0) pLo = 0; pLo &= ~3;
    int pHi = t0 + TG_COLS - 1 + 15; if (pHi > PP - 1) pHi = PP - 1;

    const float* hrow = Hest + (size_t)(n0 + lidx) * PP;
    __builtin_prefetch(hrow + pLo, 0, 1);  // global_prefetch_b8 of A band

    v8f acc[NTILES];
    #pragma unroll
    for (int jt = 0; jt < NTILES; ++jt) acc[jt] = (v8f){0,0,0,0,0,0,0,0};

    int p0 = pLo;
    // main loop: pa+1 <= p0+3 <= PP-1  =>  p0 <= PP-4 (guard-free A load)
    int pMain = (pHi <= PP - 4) ? pHi : (PP - 4);
    for (; p0 <= pMain; p0 += 4) {
        int pa = p0 + kb;                              // even -> 8B aligned
        const float2 av = *(const float2*)(hrow + pa);
        v2f a; a.x = av.x; a.y = av.y;
        int bi = VPAD_LO + t0 + lidx - pa;
        #pragma unroll
        for (int jt = 0; jt < NTILES; ++jt) {
            v2f b;
            b.x = vs[bi + 16 * jt];
            b.y = vs[bi + 16 * jt - 1];
            acc[jt] = __builtin_amdgcn_wmma_f32_16x16x4_f32(
                false, a, false, b, (short)0, acc[jt], false, false);
        }
    }
    // tail (statically unreachable for P%4==0; kept for semantic safety)
    for (; p0 <= pHi; p0 += 4) {
        int pa = p0 + kb;
        v2f a;
        a.x = (pa     < PP) ? hrow[pa]     : 0.0f;
        a.y = (pa + 1 < PP) ? hrow[pa + 1] : 0.0f;
        int bi = VPAD_LO + t0 + lidx - pa;
        #pragma unroll
        for (int jt = 0; jt < NTILES; ++jt) {
            v2f b;
            b.x = vs[bi + 16 * jt];
            b.y = vs[bi + 16 * jt - 1];
            acc[jt] = __builtin_amdgcn_wmma_f32_16x16x4_f32(
                false, a, false, b, (short)0, acc[jt], false, false);
        }
    }

    // store 8 tiles; D layout: vgpr r -> row r + 8*half, col lidx
    int mbase = half * 8;
    #pragma unroll
    for (int jt = 0; jt < NTILES; ++jt) {
        int t = t0 + jt * 16 + lidx;
        if (t >= QQ) continue;             // only trims the very last column
        #pragma unroll
        for (int r = 0; r < 8; ++r) {
            int n = n0 + mbase + r;
            int m_idx = n >> 9;            // n / K
            int k_idx = n & 511;           // n % K
            x[((size_t)(k_idx * MM + m_idx)) * QQ + t] = acc[jt][r];
        }
    }
}

// ---------------------------------------------------------------------------
// Inputs (setup_inputs order): U, H_noise, v_known, measured_toa,
//                              simulator_offset, fs, delay_max (scalars unused)
// Outputs (concat): x (K*M*Q floats), H_est (N*P floats)
// ---------------------------------------------------------------------------
extern "C" void kernel_launch(void* const* d_in, const int* in_sizes, int n_in,
                              void* d_out, int out_size, void* d_ws, size_t ws_size,
                              hipStream_t stream) {
    const float* U   = (const float*)d_in[0];
    const float* H   = (const float*)d_in[1];
    const float* v   = (const float*)d_in[2];
    const float* toa = (const float*)d_in[3];

    float* x    = (float*)d_out;
    float* Hest = (float*)d_out + (size_t)NROWS * QQ;

    float* taps = (float*)d_ws;
    int*   meta = (int*)((char*)d_ws + (size_t)NROWS * NTAP * sizeof(float));

    stage1_taps<<<NROWS / 256, 256, 0, stream>>>(U, toa, taps, meta);
    stage2_hest<<<dim3(PP / 256, NROWS), 256, 0, stream>>>(H, taps, meta, Hest);
    // 24 t-groups of 128 cols; 4 waves (t-groups) per 128-thread block
    stage3_conv_wmma<<<dim3(6, NROWS / 16), 128, 0, stream>>>(Hest, v, x);
}